// Quantum_GCNConv_79182017069407
// MI455X (gfx1250) — compile-verified
//
#include <hip/hip_runtime.h>
#include <hip/hip_bf16.h>
#include <math.h>

typedef __attribute__((ext_vector_type(2))) float v2f;
typedef __attribute__((ext_vector_type(8))) float v8f;

#define NPAIR 28
#define FDIM 256

__constant__ int c_pa[NPAIR] = {0,0,0,0,0,0,0,1,1,1,1,1,1,2,2,2,2,2,3,3,3,3,4,4,4,5,5,6};
__constant__ int c_pb[NPAIR] = {1,2,3,4,5,6,7,2,3,4,5,6,7,3,4,5,6,7,4,5,6,7,5,6,7,6,7,7};

// ---------- complex 2x2 helpers ----------
__device__ inline float2 cxmul(float2 a, float2 b) {
    return make_float2(fmaf(a.x, b.x, -a.y * b.y), fmaf(a.x, b.y, a.y * b.x));
}
__device__ inline float2 cxadd(float2 a, float2 b) { return make_float2(a.x + b.x, a.y + b.y); }
__device__ inline void mm2(const float2 A[2][2], const float2 B[2][2], float2 C[2][2]) {
    for (int i = 0; i < 2; ++i)
        for (int j = 0; j < 2; ++j)
            C[i][j] = cxadd(cxmul(A[i][0], B[0][j]), cxmul(A[i][1], B[1][j]));
}
__device__ inline void rot_x(float t, float2 R[2][2]) {
    float c = cosf(0.5f * t), s = sinf(0.5f * t);
    R[0][0] = make_float2(c, 0.f); R[0][1] = make_float2(0.f, -s);
    R[1][0] = make_float2(0.f, -s); R[1][1] = make_float2(c, 0.f);
}
__device__ inline void rot_y(float t, float2 R[2][2]) {
    float c = cosf(0.5f * t), s = sinf(0.5f * t);
    R[0][0] = make_float2(c, 0.f); R[0][1] = make_float2(-s, 0.f);
    R[1][0] = make_float2(s, 0.f); R[1][1] = make_float2(c, 0.f);
}
__device__ inline void rot_z(float t, float2 R[2][2]) {
    float c = cosf(0.5f * t), s = sinf(0.5f * t);
    R[0][0] = make_float2(c, -s); R[0][1] = make_float2(0.f, 0.f);
    R[1][0] = make_float2(0.f, 0.f); R[1][1] = make_float2(c, s);
}

// Build Mr[p] = Re(U^H diag(1,-1,1,-1) U), 28 x (4x4) real symmetric.
__global__ void k_build_mr(const float* __restrict__ qw, float* __restrict__ Mr) {
    int p = threadIdx.x;
    if (p >= NPAIR) return;
    const float* q = qw + p * 6;
    float2 R[2][2], T0[2][2], T1[2][2], A[2][2];
    rot_x(q[0], T0);
    rot_y(q[1], R); mm2(R, T0, T1);
    rot_z(q[2], R); mm2(R, T1, T0);
    rot_x(q[3], R); mm2(R, T0, T1);
    rot_y(q[4], R); mm2(R, T1, T0);
    rot_z(q[5], R); mm2(R, T0, A);
    // K = kron(A, I2)
    float2 K[4][4];
    for (int i1 = 0; i1 < 2; ++i1)
        for (int i0 = 0; i0 < 2; ++i0)
            for (int j1 = 0; j1 < 2; ++j1)
                for (int j0 = 0; j0 < 2; ++j0)
                    K[2 * i1 + i0][2 * j1 + j0] = (i0 == j0) ? A[i1][j1] : make_float2(0.f, 0.f);
    // U = CNOT01 @ CNOT10 @ K  -> row permutation of K
    const int rowsel[4] = {0, 3, 1, 2};
    float2 U[4][4];
    for (int r = 0; r < 4; ++r)
        for (int c = 0; c < 4; ++c) U[r][c] = K[rowsel[r]][c];
    const float hz[4] = {1.f, -1.f, 1.f, -1.f};
    for (int i = 0; i < 4; ++i)
        for (int j = 0; j < 4; ++j) {
            float s = 0.f;
            for (int a = 0; a < 4; ++a)
                s += hz[a] * (U[a][i].x * U[a][j].x + U[a][i].y * U[a][j].y);
            Mr[p * 16 + i * 4 + j] = s;
        }
}

// One wave per node: stage row in LDS, 28 pair-features via 4x4 quadratic forms.
__global__ __launch_bounds__(256) void k_features(const float* __restrict__ x,
                                                  const float* __restrict__ Mr,
                                                  float* __restrict__ feat, int N) {
    __shared__ float spsi[8 * FDIM];
    __shared__ float smr[NPAIR * 16];
    int w = threadIdx.x >> 5;
    int lane = threadIdx.x & 31;
    int n = blockIdx.x * 8 + w;
    for (int i = threadIdx.x; i < NPAIR * 16; i += 256) smr[i] = Mr[i];
    float ss = 0.f;
    bool active = (n < N);
    if (active) {
        const float* xr = x + (size_t)n * FDIM;
        float* ps = spsi + w * FDIM;
#pragma unroll
        for (int t = 0; t < 8; ++t) {
            float v = xr[lane + 32 * t];
            ps[lane + 32 * t] = v;
            ss = fmaf(v, v, ss);
        }
    }
    __syncthreads();
    if (!active) return;
#pragma unroll
    for (int m = 16; m >= 1; m >>= 1) ss += __shfl_xor(ss, m, 32);
    float inv = 1.0f / ss;  // feature normalizes by ||x||^2
    const float* ps = spsi + w * FDIM;
    for (int p = 0; p < NPAIR; ++p) {
        int sa = 7 - c_pa[p], sb = 7 - c_pb[p];  // bit positions (a<b -> sa>sb)
        int lo = sb, hi = sa;
        int d1 = 1 << sa, d0 = 1 << sb;
        const float* q = smr + p * 16;
        float acc = 0.f;
#pragma unroll
        for (int r = 0; r < 2; ++r) {
            int m6 = lane + 32 * r;
            int t = ((m6 >> lo) << (lo + 1)) | (m6 & ((1 << lo) - 1));
            t = ((t >> hi) << (hi + 1)) | (t & ((1 << hi) - 1));
            float w0 = ps[t], w1 = ps[t + d0], w2 = ps[t + d1], w3 = ps[t + d0 + d1];
            float t0 = q[0] * w0 + q[1] * w1 + q[2] * w2 + q[3] * w3;
            float t1 = q[4] * w0 + q[5] * w1 + q[6] * w2 + q[7] * w3;
            float t2 = q[8] * w0 + q[9] * w1 + q[10] * w2 + q[11] * w3;
            float t3 = q[12] * w0 + q[13] * w1 + q[14] * w2 + q[15] * w3;
            acc += w0 * t0 + w1 * t1 + w2 * t2 + w3 * t3;
        }
#pragma unroll
        for (int m = 16; m >= 1; m >>= 1) acc += __shfl_xor(acc, m, 32);
        if (lane == 0) feat[(size_t)n * NPAIR + p] = 0.5f * (1.0f + acc * inv);
    }
}

__global__ void k_deg_init(float* __restrict__ deg, int N) {
    int i = blockIdx.x * blockDim.x + threadIdx.x;
    if (i < N) deg[i] = 1.0f;  // self loop
}
__global__ void k_deg_acc(const int* __restrict__ ei, float* __restrict__ deg, int E) {
    int i = blockIdx.x * blockDim.x + threadIdx.x;
    if (i < E) atomicAdd(&deg[ei[E + i]], 1.0f);
}
__global__ void k_agg_init(const float* __restrict__ feat, const float* __restrict__ deg,
                           float* __restrict__ agg, int total) {
    int t = blockIdx.x * blockDim.x + threadIdx.x;
    if (t < total) agg[t] = feat[t] / deg[t / NPAIR];  // dinv[i]*1*dinv[i] self loop
}
__global__ void k_agg_edge(const int* __restrict__ ei, const float* __restrict__ feat,
                           const float* __restrict__ deg, float* __restrict__ agg, int E) {
    int e = blockIdx.x * blockDim.x + threadIdx.x;
    if (e >= E) return;
    int s = ei[e], d = ei[E + e];
    float nrm = rsqrtf(deg[s]) * rsqrtf(deg[d]);
    const float* fs = feat + (size_t)s * NPAIR;
    float* ad = agg + (size_t)d * NPAIR;
#pragma unroll
    for (int f = 0; f < NPAIR; ++f) atomicAdd(&ad[f], nrm * fs[f]);
}

__global__ void k_zero_g(float* __restrict__ G) {
    G[threadIdx.x] = 0.f;  // 1024 threads, 32x32
}

// G(32x32) += relu(agg)^T relu(agg), fp32 WMMA 16x16x4, 4 nodes/step/wave.
__global__ __launch_bounds__(256) void k_gram(const float* __restrict__ agg,
                                              float* __restrict__ G, int N, int nwaves) {
    int w = threadIdx.x >> 5, lane = threadIdx.x & 31;
    int gw = blockIdx.x * 8 + w;
    int c = lane & 15;
    int half = lane >> 4;
    v8f a00 = {}, a01 = {}, a10 = {}, a11 = {};
    for (int n0 = gw * 4; n0 < N; n0 += nwaves * 4) {
        int p = n0 + half * 2;
        int q = p + 1;
        float p0 = 0.f, p1 = 0.f, q0 = 0.f, q1 = 0.f;
        if (p < N) {
            p0 = fmaxf(agg[(size_t)p * NPAIR + c], 0.f);
            if (c + 16 < NPAIR) p1 = fmaxf(agg[(size_t)p * NPAIR + c + 16], 0.f);
        }
        if (q < N) {
            q0 = fmaxf(agg[(size_t)q * NPAIR + c], 0.f);
            if (c + 16 < NPAIR) q1 = fmaxf(agg[(size_t)q * NPAIR + c + 16], 0.f);
        }
        v2f v0 = {p0, q0};  // lane half picks K={0,1} vs K={2,3}; same image for A and B
        v2f v1 = {p1, q1};
        a00 = __builtin_amdgcn_wmma_f32_16x16x4_f32(false, v0, false, v0, (short)0, a00, false, false);
        a01 = __builtin_amdgcn_wmma_f32_16x16x4_f32(false, v0, false, v1, (short)0, a01, false, false);
        a10 = __builtin_amdgcn_wmma_f32_16x16x4_f32(false, v1, false, v0, (short)0, a10, false, false);
        a11 = __builtin_amdgcn_wmma_f32_16x16x4_f32(false, v1, false, v1, (short)0, a11, false, false);
    }
#pragma unroll
    for (int v = 0; v < 8; ++v) {
        int mrow = v + half * 8;  // C/D layout: lane<16 -> M=v, lane>=16 -> M=v+8
        atomicAdd(&G[mrow * 32 + c], a00[v]);
        atomicAdd(&G[mrow * 32 + c + 16], a01[v]);
        atomicAdd(&G[(mrow + 16) * 32 + c], a10[v]);
        atomicAdd(&G[(mrow + 16) * 32 + c + 16], a11[v]);
    }
}

__global__ __launch_bounds__(128) void k_mlp(const float* __restrict__ G,
                                             const float* __restrict__ W1, const float* __restrict__ b1,
                                             const float* __restrict__ W2, const float* __restrict__ b2,
                                             float* __restrict__ out) {
    __shared__ float sg[784];
    __shared__ float sh[128];
    int t = threadIdx.x;
    for (int i = t; i < 784; i += 128) sg[i] = G[(i / NPAIR) * 32 + (i % NPAIR)];
    __syncthreads();
    const float* wr = W1 + t * 784;
    float h = b1[t];
    for (int k = 0; k < 784; ++k) h = fmaf(wr[k], sg[k], h);
    h = fmaxf(h, 0.f);
    sh[t] = h * W2[t];
    __syncthreads();
    if (t == 0) {
        float s = b2[0];
        for (int k = 0; k < 128; ++k) s += sh[k];
        out[0] = 1.0f / (1.0f + expf(-s));
    }
}

extern "C" void kernel_launch(void* const* d_in, const int* in_sizes, int n_in,
                              void* d_out, int out_size, void* d_ws, size_t ws_size,
                              hipStream_t stream) {
    const float* x  = (const float*)d_in[0];
    const int*   ei = (const int*)d_in[1];
    const float* qw = (const float*)d_in[2];
    const float* W1 = (const float*)d_in[3];
    const float* b1 = (const float*)d_in[4];
    const float* W2 = (const float*)d_in[5];
    const float* b2 = (const float*)d_in[6];
    int N = in_sizes[0] / FDIM;
    int E = in_sizes[1] / 2;

    float* ws = (float*)d_ws;
    float* Mr   = ws;                                        // 448 floats
    float* deg  = ws + 1024;                                 // N
    float* feat = deg + ((N + 255) & ~255);                  // N*28
    float* agg  = feat + (((N * NPAIR) + 255) & ~255);       // N*28
    float* G    = agg  + (((N * NPAIR) + 255) & ~255);       // 32*32

    k_build_mr<<<1, 32, 0, stream>>>(qw, Mr);
    k_features<<<(N + 7) / 8, 256, 0, stream>>>(x, Mr, feat, N);
    k_deg_init<<<(N + 255) / 256, 256, 0, stream>>>(deg, N);
    k_deg_acc<<<(E + 255) / 256, 256, 0, stream>>>(ei, deg, E);
    k_agg_init<<<(N * NPAIR + 255) / 256, 256, 0, stream>>>(feat, deg, agg, N * NPAIR);
    k_agg_edge<<<(E + 255) / 256, 256, 0, stream>>>(ei, feat, deg, agg, E);
    k_zero_g<<<1, 1024, 0, stream>>>(G);
    const int blocks = 64;
    k_gram<<<blocks, 256, 0, stream>>>(agg, G, N, blocks * 8);
    k_mlp<<<1, 128, 0, stream>>>(G, W1, b1, W2, b2, (float*)d_out);
}